// HyperGraphModule_1236950581368
// MI455X (gfx1250) — compile-verified
//
#include <hip/hip_runtime.h>
#include <hip/hip_bf16.h>
#include <math.h>

typedef __attribute__((ext_vector_type(2))) float v2f;
typedef __attribute__((ext_vector_type(8))) float v8f;

// ---------------------------------------------------------------------------
// Y[r, 0:64] = X[r, 0:64] @ W[64,64]   (all row-major fp32)
// Block = 256 threads = 8 waves; each wave computes a 16-row x 64-col tile
// with V_WMMA_F32_16X16X4_F32: 16 k-steps x 4 n-tiles = 64 WMMAs per wave.
//
// A fragment layout (32-bit A 16x4): lanes 0-15 hold row M=lane, K={k0,k0+1};
// lanes 16-31 hold row M=lane-16, K={k0+2,k0+3}  -> per-lane global float2.
// B fragment layout (4x16): VGPR0 = row k (lanes0-15) / k+2 (lanes16-31),
// VGPR1 = k+1 / k+3 -> we pre-swizzle W in LDS as (k,k+1) float2 pairs so a
// B fragment is one ds_load_b64.
// C layout (16x16 f32, 8 VGPRs): VGPR v, lane l -> row=(l>>4)*8+v, col=l&15.
// ---------------------------------------------------------------------------
__global__ __launch_bounds__(256) void gemm_rows64_wmma(
    const float* __restrict__ X, const float* __restrict__ W,
    float* __restrict__ Y, int nrows)
{
  __shared__ v2f sW2[32 * 64];  // sW2[kpair*64 + n] = {W[2*kpair][n], W[2*kpair+1][n]}

  const int tid = threadIdx.x;
  for (int i = tid; i < 32 * 64; i += 256) {
    const int j = i >> 6;       // k-pair index
    const int n = i & 63;       // column
    v2f w;
    w.x = W[(2 * j) * 64 + n];
    w.y = W[(2 * j + 1) * 64 + n];
    sW2[i] = w;
  }
  __syncthreads();

  const int lane    = tid & 31;
  const int wave    = tid >> 5;
  const int rowBase = blockIdx.x * 128 + wave * 16;
  if (rowBase + 16 > nrows) return;   // wave-uniform guard (after barrier)

  const int m  = lane & 15;   // A-fragment row
  const int kh = lane >> 4;   // A/B k-half selector
  const int n  = lane & 15;   // B-fragment column within tile

  const float* __restrict__ xrow = X + (size_t)(rowBase + m) * 64;

  v8f acc0 = {}, acc1 = {}, acc2 = {}, acc3 = {};

#pragma unroll
  for (int kk = 0; kk < 16; ++kk) {
    const int k0 = kk * 4 + kh * 2;         // even
    v2f a = *(const v2f*)(xrow + k0);       // 8B global load

    const int kp = k0 >> 1;                  // k-pair index
    const v2f* __restrict__ brow = &sW2[kp * 64 + n];
    v2f b0 = brow[0];
    v2f b1 = brow[16];
    v2f b2 = brow[32];
    v2f b3 = brow[48];

    acc0 = __builtin_amdgcn_wmma_f32_16x16x4_f32(false, a, false, b0, (short)0, acc0, false, false);
    acc1 = __builtin_amdgcn_wmma_f32_16x16x4_f32(false, a, false, b1, (short)0, acc1, false, false);
    acc2 = __builtin_amdgcn_wmma_f32_16x16x4_f32(false, a, false, b2, (short)0, acc2, false, false);
    acc3 = __builtin_amdgcn_wmma_f32_16x16x4_f32(false, a, false, b3, (short)0, acc3, false, false);
  }

  const int crow = rowBase + (lane >> 4) * 8;
  const int ccol = lane & 15;
#pragma unroll
  for (int v = 0; v < 8; ++v) {
    float* __restrict__ yr = Y + (size_t)(crow + v) * 64 + ccol;
    yr[0]  = acc0[v];
    yr[16] = acc1[v];
    yr[32] = acc2[v];
    yr[48] = acc3[v];
  }
}

// ---------------------------------------------------------------------------
// dst[dst_idx[e], :] += src[src_idx[e], :]   for 64-wide fp32 rows.
// 16 lanes per incidence, float4 per lane, non-returning global_atomic_add_f32.
// Gathered src rows are L2-resident (<=51MB vs 192MB L2).
// ---------------------------------------------------------------------------
__global__ __launch_bounds__(256) void scatter_add64(
    const float* __restrict__ src, const int* __restrict__ src_idx,
    const int* __restrict__ dst_idx, float* __restrict__ dst, int n_inc)
{
  const int e = blockIdx.x * 16 + (threadIdx.x >> 4);
  if (e >= n_inc) return;
  const int c4 = (threadIdx.x & 15) * 4;
  const int s  = src_idx[e];
  const int d  = dst_idx[e];
  const float4 v = *(const float4*)(src + (size_t)s * 64 + c4);
  float* __restrict__ dp = dst + (size_t)d * 64 + c4;
  atomicAdd(dp + 0, v.x);
  atomicAdd(dp + 1, v.y);
  atomicAdd(dp + 2, v.z);
  atomicAdd(dp + 3, v.w);
}

// ---------------------------------------------------------------------------
// out = tanh(pre * gate), float4-vectorized. n4 = element count / 4.
// ---------------------------------------------------------------------------
__global__ __launch_bounds__(256) void gate_tanh4(
    const float* __restrict__ pre, const float* __restrict__ gate,
    float* __restrict__ out, size_t n4)
{
  const size_t i = (size_t)blockIdx.x * blockDim.x + threadIdx.x;
  if (i >= n4) return;
  const float4 p = ((const float4*)pre)[i];
  const float4 g = ((const float4*)gate)[i];
  float4 r;
  r.x = tanhf(p.x * g.x);
  r.y = tanhf(p.y * g.y);
  r.z = tanhf(p.z * g.z);
  r.w = tanhf(p.w * g.w);
  ((float4*)out)[i] = r;
}

extern "C" void kernel_launch(void* const* d_in, const int* in_sizes, int n_in,
                              void* d_out, int out_size, void* d_ws, size_t ws_size,
                              hipStream_t stream) {
  const float* node_features  = (const float*)d_in[0];
  const float* hedge_features = (const float*)d_in[1];
  const float* W_nn = (const float*)d_in[2];
  const float* W_nh = (const float*)d_in[3];
  const float* W_hh = (const float*)d_in[4];
  const float* W_hn = (const float*)d_in[5];
  const int* node_idx  = (const int*)d_in[6];
  const int* hedge_idx = (const int*)d_in[7];

  const int N   = in_sizes[0] / 64;  // 200000 nodes
  const int E   = in_sizes[1] / 64;  // 100000 hyperedges
  const int INC = in_sizes[6];       // 1.6M incidences

  float* slotA = (float*)d_ws;                 // N*64 floats
  float* slotB = slotA + (size_t)N * 64;       // N*64 floats

  float* out_node  = (float*)d_out;
  float* out_hedge = out_node + (size_t)N * 64;

  const dim3 blk(256);
  const dim3 gemmGridN((N + 127) / 128);
  const dim3 gemmGridE((E + 127) / 128);
  const dim3 scatGrid((INC + 15) / 16);

  // --- NodeConvolution ---
  // hedge_proj = hedge_features @ W_nh  -> slotA  [E,64]
  gemm_rows64_wmma<<<gemmGridE, blk, 0, stream>>>(hedge_features, W_nh, slotA, E);
  // hedge_gate = 0  -> slotB [N,64]
  hipMemsetAsync(slotB, 0, (size_t)N * 64 * sizeof(float), stream);
  // hedge_gate[node_idx] += hedge_proj[hedge_idx]
  scatter_add64<<<scatGrid, blk, 0, stream>>>(slotA, hedge_idx, node_idx, slotB, INC);
  // node_pre = node_features @ W_nn -> slotA [N,64]
  gemm_rows64_wmma<<<gemmGridN, blk, 0, stream>>>(node_features, W_nn, slotA, N);
  // new_node = tanh(node_pre * hedge_gate) -> d_out[0 : N*64)
  gate_tanh4<<<dim3(((size_t)N * 16 + 255) / 256), blk, 0, stream>>>(
      slotA, slotB, out_node, (size_t)N * 16);

  // --- HedgeConvolution ---
  // node_proj = new_node @ W_hn -> slotA [N,64]  (new_node re-read hits L2)
  gemm_rows64_wmma<<<gemmGridN, blk, 0, stream>>>(out_node, W_hn, slotA, N);
  // node_gate = 0 -> slotB [E,64]
  hipMemsetAsync(slotB, 0, (size_t)E * 64 * sizeof(float), stream);
  // node_gate[hedge_idx] += node_proj[node_idx]
  scatter_add64<<<scatGrid, blk, 0, stream>>>(slotA, node_idx, hedge_idx, slotB, INC);
  // hedge_pre = hedge_features @ W_hh -> slotA [E,64]
  gemm_rows64_wmma<<<gemmGridE, blk, 0, stream>>>(hedge_features, W_hh, slotA, E);
  // new_hedge = tanh(hedge_pre * node_gate) -> d_out[N*64 : N*64+E*64)
  gate_tanh4<<<dim3(((size_t)E * 16 + 255) / 256), blk, 0, stream>>>(
      slotA, slotB, out_hedge, (size_t)E * 16);
}